// Sparsify2D_987842478198
// MI455X (gfx1250) — compile-verified
//
#include <hip/hip_runtime.h>
#include <stdint.h>

#define HW       3136      // 56*56
#define VEC4     (HW / 4)  // 784 float4 tiles per channel
#define KTOP     940       // int(0.3 * 3136)
#define NTHREADS 256       // 8 wave32s
#define NWAVES   (NTHREADS / 32)

typedef float v4f __attribute__((ext_vector_type(4)));
typedef int   v4i __attribute__((ext_vector_type(4)));
typedef __attribute__((address_space(1))) v4i* gv4i_p;   // global int4*
typedef __attribute__((address_space(3))) v4i* lv4i_p;   // LDS int4*

// Monotone map fp32 -> u32 so that float order == unsigned order.
__device__ __forceinline__ unsigned f2key(float f) {
  unsigned u = __float_as_uint(f);
  unsigned m = (unsigned)((int)u >> 31) | 0x80000000u;
  return u ^ m;
}
__device__ __forceinline__ float key2f(unsigned k) {
  unsigned b = (k & 0x80000000u) ? (k ^ 0x80000000u) : ~k;
  return __uint_as_float(b);
}

__global__ void __launch_bounds__(NTHREADS)
sparsify2d_topk_kernel(const float* __restrict__ x, float* __restrict__ out) {
  __shared__ float    tile[HW];                  // 12544 B channel stage
  __shared__ unsigned hist[NWAVES][NTHREADS];    // per-wave32 histograms (8 KB)
  __shared__ unsigned scan[NTHREADS + 1];        // suffix sums
  __shared__ unsigned s_prefix, s_kth;

  const int    tid    = threadIdx.x;
  const int    wid    = tid >> 5;                // wave32 id, 0..7
  const size_t chbase = (size_t)blockIdx.x * HW;
  const float* src    = x + chbase;
  float*       dst    = out + chbase;

  // ---- Stage the whole channel into LDS: global is read exactly once ----
#if defined(__AMDGCN__) && __has_builtin(__builtin_amdgcn_global_load_async_to_lds_b128)
  for (int v = tid; v < VEC4; v += NTHREADS) {
    __builtin_amdgcn_global_load_async_to_lds_b128(
        (gv4i_p)(src + 4 * v),
        (lv4i_p)(tile + 4 * v),
        /*imm offset*/ 0, /*cpol*/ 0);
  }
#if __has_builtin(__builtin_amdgcn_s_wait_asynccnt)
  __builtin_amdgcn_s_wait_asynccnt(0);
#else
  asm volatile("s_wait_asynccnt 0" ::: "memory");
#endif
#else
  for (int v = tid; v < VEC4; v += NTHREADS)
    ((v4f*)tile)[v] = ((const v4f*)src)[v];
#endif
  __syncthreads();

  // ---- 4-pass MSB-first radix select: exact k-th largest of HW keys ----
  unsigned prefix = 0, kth = KTOP;
  for (int shift = 24; shift >= 0; shift -= 8) {
    const unsigned maskhi = (shift == 24) ? 0u : (0xFFFFFFFFu << (shift + 8));

    // clear all per-wave histogram copies
    #pragma unroll
    for (int w = 0; w < NWAVES; ++w) hist[w][tid] = 0u;
    __syncthreads();

    // privatized histogram: each wave32 hits its own copy (distinct banks),
    // ~8x less same-address serialization on the exponent-clustered bins
    for (int i = tid; i < HW; i += NTHREADS) {
      unsigned u = f2key(tile[i]);
      if ((u & maskhi) == prefix)                 // prefix holds bits [31:shift+8]
        atomicAdd(&hist[wid][(u >> shift) & 0xFFu], 1u);
    }
    __syncthreads();

    // fold wave copies, then suffix scan: scan[b] = # candidates with digit >= b
    unsigned h = 0;
    #pragma unroll
    for (int w = 0; w < NWAVES; ++w) h += hist[w][tid];
    scan[tid] = h;
    if (tid == 0) scan[NTHREADS] = 0u;
    __syncthreads();
    for (int off = 1; off < NTHREADS; off <<= 1) {
      unsigned add = (tid + off < NTHREADS) ? scan[tid + off] : 0u;
      __syncthreads();
      scan[tid] += add;
      __syncthreads();
    }

    const unsigned ge = scan[tid];      // digits >= tid
    const unsigned gt = scan[tid + 1];  // digits >  tid  (scan[256]==0)
    if (ge >= kth && gt < kth) {        // unique bin containing rank `kth`
      s_prefix = prefix | ((unsigned)tid << shift);
      s_kth    = kth - gt;
    }
    __syncthreads();
    prefix = s_prefix;
    kth    = s_kth;
    __syncthreads();
  }
  const float thr = key2f(prefix);      // exact k-th largest value

  // ---- Threshold from LDS, stream out non-temporally (written once) ----
  for (int v = tid; v < VEC4; v += NTHREADS) {
    v4f val = ((const v4f*)tile)[v];
    v4f c;
    c.x = (val.x >= thr) ? 1.0f : 0.0f;
    c.y = (val.y >= thr) ? 1.0f : 0.0f;
    c.z = (val.z >= thr) ? 1.0f : 0.0f;
    c.w = (val.w >= thr) ? 1.0f : 0.0f;
    val *= c;                            // comp * x (matches -0.0 semantics)
    __builtin_nontemporal_store(val, ((v4f*)dst) + v);
  }
}

extern "C" void kernel_launch(void* const* d_in, const int* in_sizes, int n_in,
                              void* d_out, int out_size, void* d_ws, size_t ws_size,
                              hipStream_t stream) {
  (void)n_in; (void)out_size; (void)d_ws; (void)ws_size;
  const float* x   = (const float*)d_in[0];
  float*       out = (float*)d_out;
  const int channels = in_sizes[0] / HW;   // 64*256 = 16384
  sparsify2d_topk_kernel<<<dim3(channels), dim3(NTHREADS), 0, stream>>>(x, out);
}